// HGNN_Encoder_70463233458752
// MI455X (gfx1250) — compile-verified
//
#include <hip/hip_runtime.h>

#define EMB 64
#define NUM_LAYERS 2
#define NUSER 30000
#define NITEM 60000
#define NGRP  2000
#define NODES (NUSER + NITEM)   // 90000

typedef __attribute__((ext_vector_type(2))) float v2f;
typedef __attribute__((ext_vector_type(8))) float v8f;

// ---------------------------------------------------------------------------
// init: d_out = concat(user_emb, item_emb, group_emb); curr = concat(user,item)
// ---------------------------------------------------------------------------
__global__ __launch_bounds__(256) void init_kernel(
    const float* __restrict__ user_emb, const float* __restrict__ item_emb,
    const float* __restrict__ group_emb, float* __restrict__ out,
    float* __restrict__ curr)
{
  long idx = (long)blockIdx.x * blockDim.x + threadIdx.x;
  const long total = (long)(NODES + NGRP) * EMB;
  if (idx >= total) return;
  long row = idx >> 6;  // / EMB
  float v;
  if (row < NUSER)      v = user_emb[idx];
  else if (row < NODES) v = item_emb[idx - (long)NUSER * EMB];
  else                  v = group_emb[idx - (long)NODES * EMB];
  out[idx] = v;
  if (row < NODES) curr[idx] = v;
}

// ---------------------------------------------------------------------------
// One K-step (K+=4) of the 16x64 tile: 1x A b64 load, 8x B b32 loads (all
// immediate offsets off two incremented base pointers), 4x V_WMMA_F32_16X16X4.
// A fragment (16x4 f32): lane<16 -> (M=lane, K=k+0..1), lane>=16 -> K=k+2..3.
// B fragment (4x16 f32): same K split across lane halves, N = lane&15.
// ---------------------------------------------------------------------------
#define GEMM_STEP(AP, BP)                                                      \
  {                                                                            \
    v2f a = *(const v2f*)(AP);                                                 \
    v2f b0, b1, b2, b3;                                                        \
    b0.x = (BP)[0];  b0.y = (BP)[64];                                          \
    b1.x = (BP)[16]; b1.y = (BP)[80];                                          \
    b2.x = (BP)[32]; b2.y = (BP)[96];                                          \
    b3.x = (BP)[48]; b3.y = (BP)[112];                                         \
    c0 = __builtin_amdgcn_wmma_f32_16x16x4_f32(false, a, false, b0, (short)0,  \
                                               c0, false, false);              \
    c1 = __builtin_amdgcn_wmma_f32_16x16x4_f32(false, a, false, b1, (short)0,  \
                                               c1, false, false);              \
    c2 = __builtin_amdgcn_wmma_f32_16x16x4_f32(false, a, false, b2, (short)0,  \
                                               c2, false, false);              \
    c3 = __builtin_amdgcn_wmma_f32_16x16x4_f32(false, a, false, b3, (short)0,  \
                                               c3, false, false);              \
  }

// ---------------------------------------------------------------------------
// C[16 x 64] tile of  A[M x K] @ B[K x 64]  using V_WMMA_F32_16X16X4_F32.
// blockIdx.x = M-tile (16 rows). 8 waves split K-steps round-robin; partial
// fp32 accumulators are summed through LDS in a fixed order (deterministic).
// If acc != nullptr:  acc[tile] += result  (fused residual add for d_out).
// C fragment: VGPR j holds (M=j, lanes 0-15) and (M=j+8, lanes 16-31).
// ---------------------------------------------------------------------------
__global__ __launch_bounds__(256) void wmma_gemm_n64(
    const float* __restrict__ A, const float* __restrict__ B,
    float* __restrict__ out, float* __restrict__ acc, int K)
{
  __shared__ float red[8 * 16 * EMB];           // 32 KB: 8 waves x 16x64 tile
  const int lane  = threadIdx.x & 31;
  const int wave  = threadIdx.x >> 5;
  const int lrow  = lane & 15;                  // M (for A) / N (for B) in tile
  const int lhalf = lane >> 4;                  // K half selector
  const long arow = (long)(blockIdx.x * 16 + lrow) * K;

  v8f c0 = {}, c1 = {}, c2 = {}, c3 = {};
  const int steps  = K >> 2;                    // K is a multiple of 4
  const int nsteps = (steps - wave + 7) >> 3;   // this wave's step count
  const int k0     = (wave << 2) + (lhalf << 1);

  const float* ap = A + arow + k0;              // A stream (HBM)
  const float* bp = B + (long)k0 * EMB + lrow;  // B rows  (L2-resident)

  int it = 0;
  for (; it + 4 <= nsteps; it += 4) {           // 16 WMMAs per 1KB of A stream
    __builtin_prefetch(ap + 640, 0, 0);         // 2.5 KB ahead on A
    GEMM_STEP(ap,      bp);
    GEMM_STEP(ap + 32, bp + 32 * EMB);
    GEMM_STEP(ap + 64, bp + 64 * EMB);
    GEMM_STEP(ap + 96, bp + 96 * EMB);
    ap += 128;
    bp += 128 * EMB;
  }
  for (; it < nsteps; ++it) {                   // per-wave remainder steps
    GEMM_STEP(ap, bp);
    ap += 32;
    bp += 32 * EMB;
  }

  // spill this wave's 16x64 partial tile to LDS
  float* r = red + wave * (16 * EMB);
#pragma unroll
  for (int j = 0; j < 8; j++) {
    const int rr = (j + (lhalf << 3)) * EMB + lrow;   // (M)*64 + N
    r[rr]      = c0[j];
    r[rr + 16] = c1[j];
    r[rr + 32] = c2[j];
    r[rr + 48] = c3[j];
  }
  __syncthreads();

  // fixed-order cross-wave reduction: each thread owns 4 consecutive outputs
  const int t = threadIdx.x;
  float4 sum = make_float4(0.f, 0.f, 0.f, 0.f);
#pragma unroll
  for (int w = 0; w < 8; w++) {
    float4 v = *(const float4*)(red + w * (16 * EMB) + t * 4);
    sum.x += v.x; sum.y += v.y; sum.z += v.z; sum.w += v.w;
  }
  const long ob = (long)blockIdx.x * (16 * EMB) + t * 4;
  if (acc) {
    float4 o = *(const float4*)(acc + ob);
    o.x += sum.x; o.y += sum.y; o.z += sum.z; o.w += sum.w;
    *(float4*)(acc + ob) = o;
  }
  *(float4*)(out + ob) = sum;
}

// ---------------------------------------------------------------------------
// Per-group attention + transform (tiny: 0.2 GFLOP total). One 64-thread
// block per group row g; thread t owns output column t.
// ---------------------------------------------------------------------------
__global__ __launch_bounds__(64) void attention_kernel(
    const float* __restrict__ user_msg, const float* __restrict__ item_msg,
    const float* __restrict__ group_emb,
    const float* __restrict__ qc_w1, const float* __restrict__ qc_b1,
    const float* __restrict__ qc_w2,
    const float* __restrict__ user_w, const float* __restrict__ user_b,
    const float* __restrict__ item_w, const float* __restrict__ item_b,
    float* __restrict__ msg_out, float* __restrict__ he_acc)
{
  __shared__ float su[EMB], si[EMB], sge[EMB], sdu[EMB], sdi[EMB],
                   sr0[EMB], sr1[EMB];
  const int g = blockIdx.x;
  const int t = threadIdx.x;
  const long gb = (long)g * EMB;
  const float u  = user_msg[gb + t];
  const float iv = item_msg[gb + t];
  const float ge = group_emb[gb + t];
  su[t] = u; si[t] = iv; sge[t] = ge;
  __syncthreads();

  // h = tanh(msg @ qc_w1 + qc_b1)
  float hu = qc_b1[t], hi = qc_b1[t];
#pragma unroll 4
  for (int k = 0; k < EMB; k++) {
    const float w = qc_w1[k * EMB + t];
    hu += su[k] * w;
    hi += si[k] * w;
  }
  hu = tanhf(hu); hi = tanhf(hi);

  // att scalar = h @ qc_w2 ; fixed-order reduction (every thread recomputes)
  const float w2 = qc_w2[t];
  sr0[t] = hu * w2; sr1[t] = hi * w2;
  __syncthreads();
  float au = 0.f, ai = 0.f;
  for (int k = 0; k < EMB; k++) { au += sr0[k]; ai += sr1[k]; }

  // softmax over the 2 attention logits
  const float mx = fmaxf(au, ai);
  const float eu = expf(au - mx), ei = expf(ai - mx);
  const float w0 = eu / (eu + ei), w1 = ei / (eu + ei);
  const float common = w0 * u + w1 * iv;
  sdu[t] = u - common; sdi[t] = iv - common;
  __syncthreads();

  // u2 = [msg-common, g_emb] @ user_w + user_b   (and same for items)
  float u2 = user_b[t], i2 = item_b[t];
#pragma unroll 4
  for (int k = 0; k < EMB; k++) {
    u2 += sdu[k] * user_w[k * EMB + t];
    i2 += sdi[k] * item_w[k * EMB + t];
  }
#pragma unroll 4
  for (int k = 0; k < EMB; k++) {
    const float gk = sge[k];
    u2 += gk * user_w[(EMB + k) * EMB + t];
    i2 += gk * item_w[(EMB + k) * EMB + t];
  }
  const float m = u2 + i2 + common;
  msg_out[gb + t] = m;
  he_acc[gb + t] += m;     // final_he accumulation (init'ed to group_emb)
}

// ---------------------------------------------------------------------------
extern "C" void kernel_launch(void* const* d_in, const int* in_sizes, int n_in,
                              void* d_out, int out_size, void* d_ws, size_t ws_size,
                              hipStream_t stream)
{
  (void)in_sizes; (void)n_in; (void)out_size; (void)ws_size;
  const float* user_emb   = (const float*)d_in[0];
  const float* item_emb   = (const float*)d_in[1];
  const float* group_emb  = (const float*)d_in[2];
  const float* user_hyper = (const float*)d_in[3];
  const float* item_hyper = (const float*)d_in[4];
  const float* full_hyper = (const float*)d_in[5];
  // d_in[6], d_in[7] are num_users / num_items (compile-time constants here)
  const float* qc_w1  = (const float*)d_in[8];
  const float* qc_b1  = (const float*)d_in[9];
  const float* qc_w2  = (const float*)d_in[10];
  const float* user_w = (const float*)d_in[11];
  const float* user_b = (const float*)d_in[12];
  const float* item_w = (const float*)d_in[13];
  const float* item_b = (const float*)d_in[14];
  float* out = (float*)d_out;

  // workspace layout (~24.6 MB of fp32)
  float* curr     = (float*)d_ws;                     // 90000 x 64
  float* user_msg = curr + (long)NODES * EMB;         //  2000 x 64
  float* item_msg = user_msg + (long)NGRP * EMB;      //  2000 x 64
  float* msg      = item_msg + (long)NGRP * EMB;      //  2000 x 64

  {
    const long total = (long)(NODES + NGRP) * EMB;
    const int blocks = (int)((total + 255) / 256);
    init_kernel<<<blocks, 256, 0, stream>>>(user_emb, item_emb, group_emb, out, curr);
  }

  for (int l = 0; l < NUM_LAYERS; l++) {
    // user_msg = user_hyper @ curr_user ; item_msg = item_hyper @ curr_item
    wmma_gemm_n64<<<NGRP / 16, 256, 0, stream>>>(
        user_hyper, curr, user_msg, nullptr, NUSER);
    wmma_gemm_n64<<<NGRP / 16, 256, 0, stream>>>(
        item_hyper, curr + (long)NUSER * EMB, item_msg, nullptr, NITEM);

    attention_kernel<<<NGRP, 64, 0, stream>>>(
        user_msg, item_msg, group_emb,
        qc_w1 + (long)l * EMB * EMB, qc_b1 + (long)l * EMB,
        qc_w2 + (long)l * EMB,
        user_w + (long)l * 2 * EMB * EMB, user_b + (long)l * EMB,
        item_w + (long)l * 2 * EMB * EMB, item_b + (long)l * EMB,
        msg, out + (long)NODES * EMB);

    // node_emb = full_hyper @ msg ; curr = node_emb ; d_out += node_emb
    wmma_gemm_n64<<<NODES / 16, 256, 0, stream>>>(
        full_hyper, msg, curr, out, NGRP);
  }
}